// Net_22239340658901
// MI455X (gfx1250) — compile-verified
//
#include <hip/hip_runtime.h>
#include <math.h>

// ---------------------------------------------------------------------------
// MI455X (gfx1250) implementation.
// All GEMMs (1536x1536x1536 adj chain, x@W1 (K=1433), S@XW, H@W2p, S@HW2p)
// go through v_wmma_f32_16x16x4_f32: one wave32 per 16x16 C tile, K stepped
// by 4, scalar remainder for K%4, fused column-bias+ReLU epilogue.
// Conv encoder stages are fused map kernels (conv3x3+relu+maxpool2 fused,
// tconv2x2 as gather, sigmoid mask fused with A-sparsity skip).
// Everything is L2-resident (hot matrices 9.4 MB << 192 MB L2).
// ---------------------------------------------------------------------------

typedef __attribute__((ext_vector_type(2))) float v2f;
typedef __attribute__((ext_vector_type(8))) float v8f;

#define NNODES 1536
#define PH1 768
#define PH2 384

// ---------------- generic f32 WMMA GEMM: C = act(A@B + bias) ---------------
// grid: (Nn/16, M/16), block: 32 (one wave per 16x16 tile). M,Nn mult of 16.
__global__ void wmma_mm_f32(const float* __restrict__ A,
                            const float* __restrict__ B,
                            float* __restrict__ C,
                            int M, int Nn, int K,
                            int lda, int ldb, int ldc,
                            const float* __restrict__ bias, int act)
{
    const int lane = threadIdx.x;       // 0..31
    const int hl   = lane >> 4;         // half: 0/1
    const int l16  = lane & 15;
    const int tm   = blockIdx.y;        // tile row
    const int tn   = blockIdx.x;        // tile col
    const int m    = tm * 16 + l16;     // A row this lane streams
    const int n    = tn * 16 + l16;     // B/C col this lane owns

    v8f acc = {};
    const int K4 = K & ~3;
    for (int k = 0; k < K4; k += 4) {
        const int ka = k + 2 * hl;
        v2f a, b;
        a.x = A[(size_t)m * lda + ka];
        a.y = A[(size_t)m * lda + ka + 1];
        b.x = B[(size_t)ka * ldb + n];
        b.y = B[(size_t)(ka + 1) * ldb + n];
        acc = __builtin_amdgcn_wmma_f32_16x16x4_f32(
            /*neg_a=*/false, a, /*neg_b=*/false, b,
            /*c_mod=*/(short)0, acc, /*reuse_a=*/false, /*reuse_b=*/false);
    }
    // scalar K remainder (e.g. K=1433): acc[v] owns C[tm*16 + v + 8*hl][n]
    for (int k = K4; k < K; ++k) {
        const float bk = B[(size_t)k * ldb + n];
#pragma unroll
        for (int v = 0; v < 8; ++v)
            acc[v] += A[(size_t)(tm * 16 + v + 8 * hl) * lda + k] * bk;
    }
    const float bn = bias ? bias[n] : 0.0f;
#pragma unroll
    for (int v = 0; v < 8; ++v) {
        float val = acc[v] + bn;
        if (act == 1) val = fmaxf(val, 0.0f);
        C[(size_t)(tm * 16 + v + 8 * hl) * ldc + n] = val;
    }
}

// ---------------- graph construction ---------------------------------------
__global__ void zero_f(float* __restrict__ p, long n) {
    long i = (long)blockIdx.x * blockDim.x + threadIdx.x;
    if (i < n) p[i] = 0.0f;
}

__global__ void scatter_edges(const int* __restrict__ ei, float* __restrict__ A, int E) {
    int i = blockIdx.x * blockDim.x + threadIdx.x;
    if (i < E) atomicAdd(&A[(size_t)ei[i] * NNODES + ei[E + i]], 1.0f);
}

// ---------------- conv encoder (fused conv3x3 + relu + maxpool2) -----------
// Input channels of the encoder are all equal to A (stack([A]) / stack([A,A])),
// so we fold cin into an effective 3x3 weight per output channel.
__global__ void conv1_pool(const float* __restrict__ A, const float* __restrict__ w,
                           const float* __restrict__ bias, float* __restrict__ out, int cin)
{
    int idx = blockIdx.x * blockDim.x + threadIdx.x;
    if (idx >= 16 * PH1 * PH1) return;
    int pw = idx % PH1; int t = idx / PH1; int ph = t % PH1; int o = t / PH1;
    float we[9];
#pragma unroll
    for (int i = 0; i < 9; ++i) {
        float s = 0.0f;
        for (int c = 0; c < cin; ++c) s += w[(o * cin + c) * 9 + i];
        we[i] = s;
    }
    const float b = bias[o];
    float mx = 0.0f;                       // all pooled values are post-ReLU (>=0)
#pragma unroll
    for (int dy = 0; dy < 2; ++dy)
#pragma unroll
        for (int dx = 0; dx < 2; ++dx) {
            int y = 2 * ph + dy, x = 2 * pw + dx;
            float s = b;
            for (int ky = -1; ky <= 1; ++ky) {
                int yy = y + ky;
                if (yy < 0 || yy >= NNODES) continue;
                for (int kx = -1; kx <= 1; ++kx) {
                    int xx = x + kx;
                    if (xx < 0 || xx >= NNODES) continue;
                    s += we[(ky + 1) * 3 + (kx + 1)] * A[(size_t)yy * NNODES + xx];
                }
            }
            mx = fmaxf(mx, fmaxf(s, 0.0f));
        }
    out[idx] = mx;
}

__global__ void conv2_pool(const float* __restrict__ in, const float* __restrict__ w,
                           const float* __restrict__ bias, float* __restrict__ out)
{
    int idx = blockIdx.x * blockDim.x + threadIdx.x;
    if (idx >= 4 * PH2 * PH2) return;
    int pw = idx % PH2; int t = idx / PH2; int ph = t % PH2; int o = t / PH2;
    const float b = bias[o];
    float mx = 0.0f;
#pragma unroll
    for (int dy = 0; dy < 2; ++dy)
#pragma unroll
        for (int dx = 0; dx < 2; ++dx) {
            int y = 2 * ph + dy, x = 2 * pw + dx;
            float s = b;
            for (int c = 0; c < 16; ++c) {
                const float* ic = in + (size_t)c * PH1 * PH1;
                const float* wc = w + (o * 16 + c) * 9;
                for (int ky = -1; ky <= 1; ++ky) {
                    int yy = y + ky;
                    if (yy < 0 || yy >= PH1) continue;
                    for (int kx = -1; kx <= 1; ++kx) {
                        int xx = x + kx;
                        if (xx < 0 || xx >= PH1) continue;
                        s += wc[(ky + 1) * 3 + (kx + 1)] * ic[(size_t)yy * PH1 + xx];
                    }
                }
            }
            mx = fmaxf(mx, fmaxf(s, 0.0f));
        }
    out[idx] = mx;
}

// tconv 2x2 stride 2 (4->16) + relu: out[o,2h+a,2w+b] = relu(sum_c p2[c,h,w]*w[c,o,a,b]+b[o])
__global__ void tconv1_relu(const float* __restrict__ p2, const float* __restrict__ w,
                            const float* __restrict__ bias, float* __restrict__ out)
{
    int idx = blockIdx.x * blockDim.x + threadIdx.x;
    if (idx >= 16 * PH1 * PH1) return;
    int x = idx % PH1; int t = idx / PH1; int y = t % PH1; int o = t / PH1;
    int h = y >> 1, a = y & 1, wd = x >> 1, bb = x & 1;
    float s = bias[o];
#pragma unroll
    for (int c = 0; c < 4; ++c)
        s += p2[((size_t)c * PH2 + h) * PH2 + wd] * w[((c * 16 + o) * 2 + a) * 2 + bb];
    out[idx] = fmaxf(s, 0.0f);
}

// attended[l][i][j] = A[i][j] * sigmoid(tconv2(u1)[l][i][j]); skip where A==0
__global__ void attended_mask(const float* __restrict__ A, const float* __restrict__ u1,
                              const float* __restrict__ t2w, const float* __restrict__ t2b,
                              float* __restrict__ out, int L)
{
    int idx = blockIdx.x * blockDim.x + threadIdx.x;
    if (idx >= L * NNODES * NNODES) return;
    int j = idx % NNODES; int t = idx / NNODES; int i = t % NNODES; int l = t / NNODES;
    const float av = A[(size_t)i * NNODES + j];
    float r = 0.0f;
    if (av != 0.0f) {
        int h = i >> 1, a = i & 1, wd = j >> 1, b = j & 1;
        float s = t2b[l];
#pragma unroll
        for (int c = 0; c < 16; ++c)
            s += u1[((size_t)c * PH1 + h) * PH1 + wd] * t2w[((c * L + l) * 2 + a) * 2 + b];
        r = av / (1.0f + __expf(-s));
    }
    out[idx] = r;
}

// ---------------- sym_norm (in place) --------------------------------------
__global__ void diag_fix(float* __restrict__ M) {
    int i = blockIdx.x * blockDim.x + threadIdx.x;
    if (i < NNODES) {
        size_t d = (size_t)i * NNODES + i;
        if (M[d] == 0.0f) M[d] = 1.0f;
    }
}

__global__ void rowsum_rsqrt(const float* __restrict__ M, float* __restrict__ dinv) {
    __shared__ float sm[256];
    const int i = blockIdx.x;
    float s = 0.0f;
    for (int j = threadIdx.x; j < NNODES; j += 256) s += M[(size_t)i * NNODES + j];
    sm[threadIdx.x] = s;
    __syncthreads();
    for (int st = 128; st > 0; st >>= 1) {
        if ((int)threadIdx.x < st) sm[threadIdx.x] += sm[threadIdx.x + st];
        __syncthreads();
    }
    if (threadIdx.x == 0) {
        float deg = sm[0];
        dinv[i] = (deg > 0.0f) ? rsqrtf(deg) : 0.0f;
    }
}

__global__ void dscale(float* __restrict__ M, const float* __restrict__ dinv) {
    long idx = (long)blockIdx.x * blockDim.x + threadIdx.x;
    if (idx >= (long)NNODES * NNODES) return;
    int i = (int)(idx / NNODES), j = (int)(idx % NNODES);
    M[idx] *= dinv[i] * dinv[j];
}

__global__ void elem_max(float* __restrict__ a, const float* __restrict__ b, long n) {
    long i = (long)blockIdx.x * blockDim.x + threadIdx.x;
    if (i < n) a[i] = fmaxf(a[i], b[i]);
}

// ---------------- tail: W2/b2 padding (7 -> 16) and log_softmax ------------
__global__ void pad_w2(const float* __restrict__ W2, const float* __restrict__ b2,
                       float* __restrict__ W2p, float* __restrict__ b2p)
{
    int idx = blockIdx.x * blockDim.x + threadIdx.x;
    if (idx < 64 * 16) {
        int r = idx >> 4, c = idx & 15;
        W2p[idx] = (c < 7) ? W2[r * 7 + c] : 0.0f;
    }
    if (idx < 16) b2p[idx] = (idx < 7) ? b2[idx] : 0.0f;
}

__global__ void log_softmax7(const float* __restrict__ OUTp, float* __restrict__ out) {
    int i = blockIdx.x * blockDim.x + threadIdx.x;
    if (i >= NNODES) return;
    float v[7], mx = -INFINITY;
#pragma unroll
    for (int j = 0; j < 7; ++j) { v[j] = OUTp[(size_t)i * 16 + j]; mx = fmaxf(mx, v[j]); }
    float s = 0.0f;
#pragma unroll
    for (int j = 0; j < 7; ++j) s += __expf(v[j] - mx);
    const float ls = logf(s) + mx;
#pragma unroll
    for (int j = 0; j < 7; ++j) out[i * 7 + j] = v[j] - ls;
}

// ---------------------------------------------------------------------------
extern "C" void kernel_launch(void* const* d_in, const int* in_sizes, int n_in,
                              void* d_out, int out_size, void* d_ws, size_t ws_size,
                              hipStream_t stream)
{
    // Inputs in setup_inputs() dict insertion order (nested dicts flattened):
    const float* x      = (const float*)d_in[0];
    const float* b0c1w  = (const float*)d_in[1];
    const float* b0c1b  = (const float*)d_in[2];
    const float* b0c2w  = (const float*)d_in[3];
    const float* b0c2b  = (const float*)d_in[4];
    const float* b0t1w  = (const float*)d_in[5];
    const float* b0t1b  = (const float*)d_in[6];
    const float* b0t2w  = (const float*)d_in[7];
    const float* b0t2b  = (const float*)d_in[8];
    const float* b1c1w  = (const float*)d_in[9];
    const float* b1c1b  = (const float*)d_in[10];
    const float* b1c2w  = (const float*)d_in[11];
    const float* b1c2b  = (const float*)d_in[12];
    const float* b1t1w  = (const float*)d_in[13];
    const float* b1t1b  = (const float*)d_in[14];
    const float* b1t2w  = (const float*)d_in[15];
    const float* b1t2b  = (const float*)d_in[16];
    const float* W1     = (const float*)d_in[17];
    const float* b1g    = (const float*)d_in[18];
    const float* W2     = (const float*)d_in[19];
    const float* b2g    = (const float*)d_in[20];
    const int*   edge   = (const int*)d_in[n_in - 1];
    const int    E      = in_sizes[n_in - 1] / 2;

    const size_t NN = (size_t)NNODES * NNODES;
    float* ws   = (float*)d_ws;
    float* A_   = ws;                  // NN
    float* att0 = ws + 1 * NN;         // NN  (b1 attended[0]; contiguous with att1)
    float* att1 = ws + 2 * NN;         // NN  (b1 attended[1])
    float* adj1 = ws + 3 * NN;         // NN  (b0 result; later max & final S)
    float* adj2 = ws + 4 * NN;         // NN  (b1 result / big matmul out)
    float* p1   = ws + 5 * NN;         // 16*768*768 = 4*NN
    float* u1   = ws + 9 * NN;         // 16*768*768 = 4*NN
    float* p2   = ws + 13 * NN;        // 4*384*384
    float* dinv = p2 + (size_t)4 * PH2 * PH2;
    float* XW   = dinv + NNODES;       // 1536*64
    float* Hh   = XW + (size_t)NNODES * 64;
    float* W2p  = Hh + (size_t)NNODES * 64;  // 64*16
    float* b2gp = W2p + 64 * 16;             // 16
    float* HW2p = b2gp + 16;                 // 1536*16
    float* OUTp = HW2p + (size_t)NNODES * 16;

    const int T = 256;
    auto blocks = [&](long n) { return (unsigned)((n + T - 1) / T); };
    auto sym_norm = [&](float* M) {
        diag_fix<<<blocks(NNODES), T, 0, stream>>>(M);
        rowsum_rsqrt<<<NNODES, T, 0, stream>>>(M, dinv);
        dscale<<<blocks((long)NN), T, 0, stream>>>(M, dinv);
    };

    // 1) dense adjacency
    zero_f<<<blocks((long)NN), T, 0, stream>>>(A_, (long)NN);
    scatter_edges<<<blocks(E), T, 0, stream>>>(edge, A_, E);

    // 2) block b0 (cin=1, L=1) -> adj1
    conv1_pool<<<blocks(16L * PH1 * PH1), T, 0, stream>>>(A_, b0c1w, b0c1b, p1, 1);
    conv2_pool<<<blocks(4L * PH2 * PH2), T, 0, stream>>>(p1, b0c2w, b0c2b, p2);
    tconv1_relu<<<blocks(16L * PH1 * PH1), T, 0, stream>>>(p2, b0t1w, b0t1b, u1);
    attended_mask<<<blocks((long)NN), T, 0, stream>>>(A_, u1, b0t2w, b0t2b, adj1, 1);
    sym_norm(adj1);                    // adj1 = sym_norm(I @ attended[0])

    // 3) block b1 (cin=2, L=2) -> adj2
    conv1_pool<<<blocks(16L * PH1 * PH1), T, 0, stream>>>(A_, b1c1w, b1c1b, p1, 2);
    conv2_pool<<<blocks(4L * PH2 * PH2), T, 0, stream>>>(p1, b1c2w, b1c2b, p2);
    tconv1_relu<<<blocks(16L * PH1 * PH1), T, 0, stream>>>(p2, b1t1w, b1t1b, u1);
    attended_mask<<<blocks(2L * NN), T, 0, stream>>>(A_, u1, b1t2w, b1t2b, att0, 2);
    sym_norm(att0);                    // after meta-path hop 0
    // big 1536^3 fp32 WMMA matmul: adj2 = sym_norm(att0) @ att1
    wmma_mm_f32<<<dim3(NNODES / 16, NNODES / 16), 32, 0, stream>>>(
        att0, att1, adj2, NNODES, NNODES, NNODES, NNODES, NNODES, NNODES, nullptr, 0);
    sym_norm(adj2);

    // 4) coalesce max; 5) final renormalized propagation matrix S (in adj1)
    elem_max<<<blocks((long)NN), T, 0, stream>>>(adj1, adj2, (long)NN);
    sym_norm(adj1);

    // 6) GCN layer 1: H = relu(S @ (x @ W1) + b1g)
    wmma_mm_f32<<<dim3(64 / 16, NNODES / 16), 32, 0, stream>>>(
        x, W1, XW, NNODES, 64, 1433, 1433, 64, 64, nullptr, 0);
    wmma_mm_f32<<<dim3(64 / 16, NNODES / 16), 32, 0, stream>>>(
        adj1, XW, Hh, NNODES, 64, NNODES, NNODES, 64, 64, b1g, 1);

    // 7) GCN layer 2 (N=7 padded to 16): OUT = S @ (H @ W2p) + b2gp
    pad_w2<<<blocks(64 * 16), T, 0, stream>>>(W2, b2g, W2p, b2gp);
    wmma_mm_f32<<<dim3(16 / 16, NNODES / 16), 32, 0, stream>>>(
        Hh, W2p, HW2p, NNODES, 16, 64, 64, 16, 16, nullptr, 0);
    wmma_mm_f32<<<dim3(16 / 16, NNODES / 16), 32, 0, stream>>>(
        adj1, HW2p, OUTp, NNODES, 16, NNODES, NNODES, 16, 16, b2gp, 0);

    // 8) log_softmax over the 7 real classes
    log_softmax7<<<blocks(NNODES), T, 0, stream>>>(OUTp, (float*)d_out);
}